// DiagGcn_70884140253773
// MI455X (gfx1250) — compile-verified
//
#include <hip/hip_runtime.h>
#include <cstdint>

#define D_DIM 128
#define BLOCK 256
#define CHUNK 256                      // edges staged per block per iteration
#define WAVES_PER_BLOCK (BLOCK / 32)   // 8 waves (wave32)
#define EDGES_PER_WAVE (CHUNK / WAVES_PER_BLOCK) // 32

// ---- CDNA5 async global->LDS copy (ASYNCcnt path) -------------------------
// Generic pointers to LDS on amdgcn carry the LDS byte offset in the low 32
// bits (shared aperture in the high bits), so truncation gives the DS offset.
__device__ __forceinline__ uint32_t lds_off_of(const void* p) {
  return (uint32_t)(uintptr_t)p;
}

__device__ __forceinline__ void async_ld_b32(uint32_t lds_off, const void* gptr) {
  asm volatile("global_load_async_to_lds_b32 %0, %1, off"
               :
               : "v"(lds_off), "v"((uint64_t)(uintptr_t)gptr)
               : "memory");
}

__device__ __forceinline__ void wait_async_zero() {
  asm volatile("s_wait_asynccnt 0" ::: "memory");
}

// ---- main edge kernel: wave-per-edge, lane = float4 slice -----------------
// (First in the file so the disasm snippet shows this kernel.)
__global__ __launch_bounds__(BLOCK) void diaggcn_edge_kernel(
    const float* __restrict__ Vs,    // [N, 128] sender embeddings
    const float* __restrict__ Vt,    // [401, 128] relation diagonals
    const float* __restrict__ Bm,    // [128] bias
    const float* __restrict__ wgt,   // [E] incidence weights
    const int*   __restrict__ snd,   // [E]
    const int*   __restrict__ typ,   // [E]
    const int*   __restrict__ rcv,   // [E]
    float*       __restrict__ out,   // [N, 128]
    int nEdges, int nChunks)
{
  __shared__ int   sSnd[2][CHUNK];
  __shared__ int   sTyp[2][CHUNK];
  __shared__ int   sRcv[2][CHUNK];
  __shared__ float sW  [2][CHUNK];

  const int tid  = threadIdx.x;
  const int lane = tid & 31;
  const int wv   = tid >> 5;

  // Bias slice for this lane (loop-invariant, lives in 4 VGPRs).
  const float4 b4 = *(const float4*)(Bm + lane * 4);

  auto stage = [&](int chunk, int buf) {
    const int e = chunk * CHUNK + tid;
    if (e < nEdges) {
      async_ld_b32(lds_off_of(&sSnd[buf][tid]), snd + e);
      async_ld_b32(lds_off_of(&sTyp[buf][tid]), typ + e);
      async_ld_b32(lds_off_of(&sRcv[buf][tid]), rcv + e);
      async_ld_b32(lds_off_of(&sW  [buf][tid]), wgt + e);
    }
  };

  int c   = blockIdx.x;
  int buf = 0;
  if (c < nChunks) stage(c, 0);
  wait_async_zero();
  __syncthreads();

  for (; c < nChunks; c += gridDim.x) {
    const int cn = c + gridDim.x;
    if (cn < nChunks) stage(cn, buf ^ 1);   // prefetch next chunk's metadata

    const int ebase  = c * CHUNK + wv * EDGES_PER_WAVE;
    const int ecount = min(EDGES_PER_WAVE, nEdges - ebase);

    for (int i = 0; i < ecount; ++i) {
      const int el = wv * EDGES_PER_WAVE + i;
      const int s  = sSnd[buf][el];   // wave-uniform LDS broadcast reads
      const int t  = sTyp[buf][el];
      const int r  = sRcv[buf][el];
      const float ew = sW[buf][el];

      // Prefetch the sender row two edges ahead (gfx1250 global_prefetch).
      if (i + 2 < ecount) {
        const int s2 = sSnd[buf][el + 2];
        __builtin_prefetch(Vs + (size_t)s2 * D_DIM + lane * 4, 0, 0);
      }

      const float4 a = *(const float4*)(Vs + (size_t)s * D_DIM + lane * 4);
      const float4 d = *(const float4*)(Vt + (size_t)t * D_DIM + lane * 4);

      float4 m;
      m.x = fmaxf(fmaf(a.x, d.x, b4.x), 0.f) * ew;
      m.y = fmaxf(fmaf(a.y, d.y, b4.y), 0.f) * ew;
      m.z = fmaxf(fmaf(a.z, d.z, b4.z), 0.f) * ew;
      m.w = fmaxf(fmaf(a.w, d.w, b4.w), 0.f) * ew;

      // Guaranteed hardware fp32 atomics (no-return -> STOREcnt), resolved in
      // L2 at device scope; one 64-bit address + immediate offsets 0/4/8/12.
      const uint64_t oaddr =
          (uint64_t)(uintptr_t)(out + (size_t)r * D_DIM + lane * 4);
      asm volatile("global_atomic_add_f32 %0, %1, off scope:SCOPE_DEV"
                   :: "v"(oaddr), "v"(m.x) : "memory");
      asm volatile("global_atomic_add_f32 %0, %1, off offset:4 scope:SCOPE_DEV"
                   :: "v"(oaddr), "v"(m.y) : "memory");
      asm volatile("global_atomic_add_f32 %0, %1, off offset:8 scope:SCOPE_DEV"
                   :: "v"(oaddr), "v"(m.z) : "memory");
      asm volatile("global_atomic_add_f32 %0, %1, off offset:12 scope:SCOPE_DEV"
                   :: "v"(oaddr), "v"(m.w) : "memory");
    }

    wait_async_zero();   // next chunk's metadata landed
    __syncthreads();
    buf ^= 1;
  }
}

// ---- zero the output (harness poisons d_out; must re-zero every launch) ---
__global__ __launch_bounds__(BLOCK) void diaggcn_zero_kernel(float4* __restrict__ p, int n4) {
  int i = blockIdx.x * BLOCK + threadIdx.x;
  if (i < n4) p[i] = make_float4(0.f, 0.f, 0.f, 0.f);
}

extern "C" void kernel_launch(void* const* d_in, const int* in_sizes, int n_in,
                              void* d_out, int out_size, void* d_ws, size_t ws_size,
                              hipStream_t stream) {
  const float* Vs  = (const float*)d_in[0];  // V_proj_sender [N*128]
  const float* Vt  = (const float*)d_in[1];  // V_types [401*128]
  const float* Bm  = (const float*)d_in[2];  // B_message [128]
  const float* wgt = (const float*)d_in[3];  // inc_weights [E]
  const int*   snd = (const int*)d_in[4];    // sender_idx [E]
  const int*   typ = (const int*)d_in[5];    // type_idx [E]
  const int*   rcv = (const int*)d_in[6];    // recv_idx [E]
  float* out = (float*)d_out;

  const int nEdges = in_sizes[3];
  const int n4     = out_size / 4;

  // Zero the (poisoned) output accumulator.
  diaggcn_zero_kernel<<<(n4 + BLOCK - 1) / BLOCK, BLOCK, 0, stream>>>(
      (float4*)out, n4);

  const int nChunks = (nEdges + CHUNK - 1) / CHUNK;
  int grid = nChunks < 1024 ? nChunks : 1024;  // grid-stride over chunks
  if (grid < 1) grid = 1;

  diaggcn_edge_kernel<<<grid, BLOCK, 0, stream>>>(
      Vs, Vt, Bm, wgt, snd, typ, rcv, out, nEdges, nChunks);
}